// _MatrixDecomposition2DBase_8589934834
// MI455X (gfx1250) — compile-verified
//
#include <hip/hip_runtime.h>
#include <math.h>

// ---------------------------------------------------------------------------
// NMF matrix decomposition on x[8,512,4096], rank 64 (see reference).
// All GEMMs run through V_WMMA_F32_16X16X4_F32 (f32 in / f32 acc, matches
// reference numerics). Whole working set (~80 MB) is L2-resident on MI455X.
// Skinny GEMMs (64x64 Gram matrices, 512x64 basesnum) use deterministic
// split-K: per-chunk partials + fixed-order reduction (no float atomics).
// ---------------------------------------------------------------------------
namespace {
constexpr int Bn = 8;
constexpr int Dd = 512;
constexpr int Nn = 4096;
constexpr int Rr = 64;
constexpr float EPS = 1e-6f;
constexpr float INV_T = 100.0f;
}

typedef float v2f __attribute__((ext_vector_type(2)));
typedef float v8f __attribute__((ext_vector_type(8)));

// ---------------------------------------------------------------------------
// bases[b,:,r] = bases0[b,:,r] / max(||bases0[b,:,r]||_2, 1e-12)
// One wave per (b,r) column of 512 elements (stride R).
// ---------------------------------------------------------------------------
__global__ __launch_bounds__(128) void normalize_bases(const float* __restrict__ b0,
                                                       float* __restrict__ bases) {
  const int wave = threadIdx.x >> 5;
  const int lane = threadIdx.x & 31;
  const int col  = blockIdx.x * 4 + wave;      // 0 .. B*R-1
  const int b = col >> 6;
  const int r = col & 63;
  const float* src = b0    + (long)b * Dd * Rr + r;
  float*       dst = bases + (long)b * Dd * Rr + r;

  float s = 0.f;
  for (int d = lane; d < Dd; d += 32) {
    float v = src[(long)d * Rr];
    s += v * v;
  }
  for (int m = 16; m; m >>= 1) s += __shfl_xor(s, m, 32);
  const float inv = 1.f / fmaxf(sqrtf(s), 1e-12f);
  for (int d = lane; d < Dd; d += 32) dst[(long)d * Rr] = src[(long)d * Rr] * inv;
}

// ---------------------------------------------------------------------------
// dst[b][i] = sum_{c < nchunks} src[b][c][i]   (fixed order -> deterministic)
// ---------------------------------------------------------------------------
__global__ __launch_bounds__(256) void reduce_chunks(const float* __restrict__ src,
                                                     float* __restrict__ dst,
                                                     int plane, int nchunks) {
  const int idx = blockIdx.x * 256 + threadIdx.x;
  if (idx >= Bn * plane) return;
  const int b = idx / plane;
  const int i = idx - b * plane;
  const float* p = src + (long)b * nchunks * plane + i;
  float s = 0.f;
  for (int c = 0; c < nchunks; ++c) s += p[(long)c * plane];
  dst[idx] = s;
}

// ---------------------------------------------------------------------------
// Templated f32 WMMA GEMM: per batch b = blockIdx.z,
//   acc(m,n) = sum_k A(m,k)*B(k,n)   for a 64x64 tile at (m0, n0)
//   A(m,k) = A[TA ? k*lda+m : m*lda+k],  B(k,n) = B[TB ? n*ldb+k : k*ldb+n]
// Block = 128 threads = 4 waves; wave w owns rows [m0+16w, m0+16w+16) x 64 cols.
//
// Fragment layout (ISA 7.12.2, wave32):
//   A 16x4 : lane l -> row m=l&15, a.x=A(m, kb), a.y=A(m, kb+1), kb=2*(l>>4)
//   B 4x16 : lane l -> col n=l&15, b.x=B(kb, n), b.y=B(kb+1, n)
//   C 16x16: vgpr i, lane l -> row i+8*(l>>4), col l&15
//
// SPLIT == 1: blockIdx.y selects a K-chunk of size Kc; the 64-col (N==R)
//   partial tile is stored at C + b*sC + chunk*M*64 (MODE must be 0).
// MODE 0: C = acc
// MODE 1: C = C * Np / (acc + EPS)   (in-place NMF update; each wave reads only
//         its own 16 rows of A==C and the full K=64 contraction is in-block)
// MODE 2: C = softmax(INV_T * acc) over 64 cols (row lives in one wave; reduce
//         within each 16-lane half via shfl_xor 8,4,2,1)
// ---------------------------------------------------------------------------
template <int TA, int TB, int MODE, int SPLIT>
__global__ __launch_bounds__(128) void wmma_gemm_f32(
    const float* Ap, const float* __restrict__ Bp, float* Cp,
    const float* __restrict__ Np,
    int M, int Kc, int lda, int ldb, int ldc,
    long sA, long sB, long sC) {
  const int wave = threadIdx.x >> 5;
  const int lane = threadIdx.x & 31;
  const int b    = blockIdx.z;
  const int m0   = blockIdx.x * 64 + wave * 16;
  const int n0   = SPLIT ? 0 : blockIdx.y * 64;
  const int klo  = SPLIT ? blockIdx.y * Kc : 0;
  if (m0 >= M) return;

  const float* A = Ap + (long)b * sA;
  const float* B = Bp + (long)b * sB;
  float*       C = Cp + (long)b * sC + (SPLIT ? (long)blockIdx.y * M * 64 : 0L);

  const int half = lane >> 4;     // 0 or 1
  const int l16  = lane & 15;
  const int kb   = half * 2;

  v8f acc[4];
#pragma unroll
  for (int c = 0; c < 4; ++c)
#pragma unroll
    for (int j = 0; j < 8; ++j) acc[c][j] = 0.0f;

#pragma unroll 4
  for (int kk = 0; kk < Kc; kk += 4) {
    const int k0 = klo + kk;
    v2f a;
    if (TA) {
      const float* p = A + (long)(k0 + kb) * lda + (m0 + l16);
      a.x = p[0];
      a.y = p[lda];
    } else {
      // (k0+kb) even, lda even -> 8B-aligned contiguous pair
      a = *(const v2f*)(A + (long)(m0 + l16) * lda + (k0 + kb));
    }
    v2f bf[4];
#pragma unroll
    for (int c = 0; c < 4; ++c) {
      const int n = n0 + c * 16 + l16;
      if (TB) {
        bf[c] = *(const v2f*)(B + (long)n * ldb + (k0 + kb));
      } else {
        const float* p = B + (long)(k0 + kb) * ldb + n;
        bf[c].x = p[0];
        bf[c].y = p[ldb];
      }
    }
#pragma unroll
    for (int c = 0; c < 4; ++c) {
      acc[c] = __builtin_amdgcn_wmma_f32_16x16x4_f32(
          false, a, false, bf[c], (short)0, acc[c], false, false);
    }
  }

  if (MODE == 0) {
#pragma unroll
    for (int i = 0; i < 8; ++i) {
      const int m = m0 + i + 8 * half;
#pragma unroll
      for (int c = 0; c < 4; ++c)
        C[(long)m * ldc + (n0 + c * 16 + l16)] = acc[c][i];
    }
  } else if (MODE == 1) {
    const float* Nb = Np + (long)b * sC;
#pragma unroll
    for (int i = 0; i < 8; ++i) {
      const int m = m0 + i + 8 * half;
#pragma unroll
      for (int c = 0; c < 4; ++c) {
        const long idx = (long)m * ldc + (n0 + c * 16 + l16);
        C[idx] = C[idx] * Nb[idx] / (acc[c][i] + EPS);
      }
    }
  } else {  // MODE 2: row softmax over 64 columns (N tile == full R)
#pragma unroll
    for (int i = 0; i < 8; ++i) {
      const int m = m0 + i + 8 * half;
      float v[4];
      float mx = -3.402823466e38f;
#pragma unroll
      for (int c = 0; c < 4; ++c) { v[c] = acc[c][i] * INV_T; mx = fmaxf(mx, v[c]); }
      for (int s = 8; s; s >>= 1) mx = fmaxf(mx, __shfl_xor(mx, s, 32));
      float sum = 0.f;
#pragma unroll
      for (int c = 0; c < 4; ++c) { v[c] = expf(v[c] - mx); sum += v[c]; }
      for (int s = 8; s; s >>= 1) sum += __shfl_xor(sum, s, 32);
      const float inv = 1.f / sum;
#pragma unroll
      for (int c = 0; c < 4; ++c)
        C[(long)m * ldc + (n0 + c * 16 + l16)] = v[c] * inv;
    }
  }
}

// ---------------------------------------------------------------------------
// Host side: stream-ordered launches only (graph-capture safe).
// ---------------------------------------------------------------------------
template <int TA, int TB, int MODE, int SPLIT>
static inline void launch_gemm(const float* A, const float* Bm, float* C,
                               const float* Np, int M, int gy, int Kc,
                               int lda, int ldb, int ldc,
                               long sA, long sB, long sC, hipStream_t stream) {
  dim3 grid(M / 64, gy, Bn);  // gy = N/64 (dense) or nchunks (split-K)
  wmma_gemm_f32<TA, TB, MODE, SPLIT><<<grid, dim3(128), 0, stream>>>(
      A, Bm, C, Np, M, Kc, lda, ldb, ldc, sA, sB, sC);
}

extern "C" void kernel_launch(void* const* d_in, const int* in_sizes, int n_in,
                              void* d_out, int out_size, void* d_ws, size_t ws_size,
                              hipStream_t stream) {
  const float* x      = (const float*)d_in[0];   // [B][D][N]
  const float* bases0 = (const float*)d_in[1];   // [B][D][R]
  float* out = (float*)d_out;                    // [B][D][N]

  float* ws       = (float*)d_ws;
  float* bases    = ws;                                  // B*D*R      (1 MB)
  float* coef     = bases    + (size_t)Bn * Dd * Rr;     // B*N*R      (8 MB)
  float* coefnum  = coef     + (size_t)Bn * Nn * Rr;     // B*N*R      (8 MB)
  float* basesnum = coefnum  + (size_t)Bn * Nn * Rr;     // B*D*R      (1 MB)
  float* rtr      = basesnum + (size_t)Bn * Dd * Rr;     // B*R*R      (128 KB)
  float* partial  = rtr      + (size_t)Bn * Rr * Rr;     // B*8*D*R    (8 MB max)

  const long sX  = (long)Dd * Nn;
  const long sBa = (long)Dd * Rr;
  const long sCo = (long)Nn * Rr;
  const long sR  = (long)Rr * Rr;

  auto reduce = [&](const float* src, float* dst, int plane, int nchunks) {
    const int total = Bn * plane;
    reduce_chunks<<<dim3((total + 255) / 256), dim3(256), 0, stream>>>(
        src, dst, plane, nchunks);
  };

  // bases = colnormalize(bases0)
  normalize_bases<<<dim3(Bn * Rr / 4), dim3(128), 0, stream>>>(bases0, bases);

  // coef = softmax(INV_T * x^T bases)   [N x R, K = D]
  launch_gemm<1, 0, 2, 0>(x, bases, coef, nullptr, Nn, 1, Dd, Nn, Rr, Rr,
                          sX, sBa, sCo, stream);

  auto coef_update = [&]() {
    // coefnum = x^T bases               [N x R, K = D]  (512 blocks, dense)
    launch_gemm<1, 0, 0, 0>(x, bases, coefnum, nullptr, Nn, 1, Dd, Nn, Rr, Rr,
                            sX, sBa, sCo, stream);
    // rtr = bases^T bases               [R x R, K = D]  split-K: 4 x 128
    launch_gemm<1, 0, 0, 1>(bases, bases, partial, nullptr, Rr, 4, 128,
                            Rr, Rr, Rr, sBa, sBa, 4L * Rr * Rr, stream);
    reduce(partial, rtr, Rr * Rr, 4);
    // coef = coef * coefnum / (coef@rtr + eps)   [N x R, K = R]
    launch_gemm<0, 0, 1, 0>(coef, rtr, coef, coefnum, Nn, 1, Rr, Rr, Rr, Rr,
                            sCo, sR, sCo, stream);
  };
  auto bases_update = [&]() {
    // basesnum = x coef                 [D x R, K = N]  split-K: 8 x 512
    launch_gemm<0, 0, 0, 1>(x, coef, partial, nullptr, Dd, 8, 512,
                            Nn, Rr, Rr, sX, sCo, 8L * Dd * Rr, stream);
    reduce(partial, basesnum, Dd * Rr, 8);
    // rtr = coef^T coef                 [R x R, K = N]  split-K: 16 x 256
    launch_gemm<1, 0, 0, 1>(coef, coef, partial, nullptr, Rr, 16, 256,
                            Rr, Rr, Rr, sCo, sCo, 16L * Rr * Rr, stream);
    reduce(partial, rtr, Rr * Rr, 16);
    // bases = bases * basesnum / (bases@rtr + eps)   [D x R, K = R]
    launch_gemm<0, 0, 1, 0>(bases, rtr, bases, basesnum, Dd, 1, Rr, Rr, Rr, Rr,
                            sBa, sR, sBa, stream);
  };

  for (int it = 0; it < 7; ++it) { coef_update(); bases_update(); }
  coef_update();  // compute_coef: one more coef update

  // out = bases @ coef^T                [D x N, K = R]  (4096 blocks, dense)
  launch_gemm<0, 1, 0, 0>(bases, coef, out, nullptr, Dd, Nn / 64, Rr,
                          Rr, Rr, Nn, sBa, sCo, sX, stream);
}